// SEBlock_17033840296532
// MI455X (gfx1250) — compile-verified
//
#include <hip/hip_runtime.h>
#include <hip/hip_bf16.h>

// SE block: x[32,256,56,56] fp32.
//   s = mean_{H,W}(x)            -> [32,256]
//   h = silu(s @ w1^T + b1)      -> [32,16]
//   g = sigmoid(h @ w2^T + b2)   -> [32,256]
//   out = x * g[:, :, None, None]
//
// Memory-bound: ~308MB cold / ~206MB with x resident in 192MB L2.
// Excitation matmuls done with V_WMMA_F32_16X16X4_F32 (exact f32 math).

#define SE_B   32
#define SE_C   256
#define SE_CM  16
#define SE_HW  3136      // 56*56
#define SE_HW4 784       // 3136/4 float4s per channel

typedef __attribute__((ext_vector_type(2))) float v2f;
typedef __attribute__((ext_vector_type(8))) float v8f;

// ---------------- Kernel 1: squeeze (per-channel mean) ----------------
// One block per (b,c); 3136 contiguous floats = 784 float4s, 256 threads.
__global__ __launch_bounds__(256) void se_reduce(const float* __restrict__ x,
                                                 float* __restrict__ s) {
    const int bc = blockIdx.x;                       // b*C + c
    const float4* xv = (const float4*)(x + (size_t)bc * SE_HW);
    const int t = threadIdx.x;

    float sum = 0.0f;
#pragma unroll
    for (int i = 0; i < 3; ++i) {                    // 3*256 = 768 float4s
        float4 v = xv[t + i * 256];
        sum += v.x + v.y + v.z + v.w;
    }
    if (t < SE_HW4 - 768) {                          // remaining 16 float4s
        float4 v = xv[t + 768];
        sum += v.x + v.y + v.z + v.w;
    }

    // wave32 reduction
#pragma unroll
    for (int off = 16; off > 0; off >>= 1)
        sum += __shfl_xor(sum, off, 32);

    __shared__ float wsum[8];
    const int wave = t >> 5;
    if ((t & 31) == 0) wsum[wave] = sum;
    __syncthreads();
    if (t == 0) {
        float tot = 0.0f;
#pragma unroll
        for (int i = 0; i < 8; ++i) tot += wsum[i];
        s[bc] = tot * (1.0f / (float)SE_HW);
    }
}

// ---------------- Kernel 2: excitation via f32 WMMA ----------------
// Single block, 16 waves (512 threads).
// Phase 1: waves 0..1 compute h[32,16] = silu(s @ w1^T + b1)   (K=256 loop)
// Phase 2: 32 output tiles of g[32,256] = sigmoid(h @ w2^T + b2), 2 per wave.
__global__ __launch_bounds__(512) void se_excite(const float* __restrict__ s,
                                                 const float* __restrict__ w1,
                                                 const float* __restrict__ b1,
                                                 const float* __restrict__ w2,
                                                 const float* __restrict__ b2,
                                                 float* __restrict__ g) {
    __shared__ float h_lds[SE_B * SE_CM];            // 32x16 = 2KB

    const int lane = threadIdx.x & 31;
    const int wave = threadIdx.x >> 5;
    const int l15  = lane & 15;                      // N (or M-row) within tile
    const int khi  = (lane >> 4) << 1;               // 0 for lanes 0-15, 2 for 16-31
    const int rrow = (lane >> 4) << 3;               // D-row offset: 0 or 8

    // ---- Phase 1: h = silu(s @ w1^T + b1), M tiles handled by waves 0,1 ----
    if (wave < 2) {
        const int m0 = wave * 16;
        v8f acc = {};
        for (int k0 = 0; k0 < SE_C; k0 += 4) {
            v2f a, b;
            // A[m,k] = s[m0+l15, k]  (16x4 f32 layout per ISA)
            const int arow = (m0 + l15) * SE_C + k0 + khi;
            a.x = s[arow];
            a.y = s[arow + 1];
            // B[k,n] = w1^T[k, n] = w1[n, k],  n = l15
            const int brow = l15 * SE_C + k0 + khi;
            b.x = w1[brow];
            b.y = w1[brow + 1];
            acc = __builtin_amdgcn_wmma_f32_16x16x4_f32(
                false, a, false, b, (short)0, acc, false, false);
        }
        // D layout: VGPR r holds row m0 + r + rrow, col l15
        const float bias = b1[l15];
#pragma unroll
        for (int r = 0; r < 8; ++r) {
            const int row = m0 + r + rrow;
            const float v = acc[r] + bias;
            const float sig = 1.0f / (1.0f + __expf(-v));
            h_lds[row * SE_CM + l15] = v * sig;       // silu
        }
    }
    __syncthreads();

    // ---- Phase 2: g = sigmoid(h @ w2^T + b2); 2x16 grid of 16x16 tiles ----
#pragma unroll
    for (int i = 0; i < 2; ++i) {
        const int tile = wave * 2 + i;               // 0..31
        const int m0 = (tile >> 4) * 16;             // 0 or 16
        const int n0 = (tile & 15) * 16;             // 0..240
        v8f acc = {};
#pragma unroll
        for (int k0 = 0; k0 < SE_CM; k0 += 4) {
            v2f a, b;
            // A[m,k] = h[m0+l15, k]
            const int arow = (m0 + l15) * SE_CM + k0 + khi;
            a.x = h_lds[arow];
            a.y = h_lds[arow + 1];
            // B[k,n] = w2[n0+n, k],  w2 is [C, C_MID]
            const int brow = (n0 + l15) * SE_CM + k0 + khi;
            b.x = w2[brow];
            b.y = w2[brow + 1];
            acc = __builtin_amdgcn_wmma_f32_16x16x4_f32(
                false, a, false, b, (short)0, acc, false, false);
        }
        const int col = n0 + l15;
        const float bias = b2[col];
#pragma unroll
        for (int r = 0; r < 8; ++r) {
            const int row = m0 + r + rrow;
            const float v = acc[r] + bias;
            g[row * SE_C + col] = 1.0f / (1.0f + __expf(-v));
        }
    }
}

// ---------------- Kernel 3: scale (out = x * g) ----------------
// One block per (b,c); g value is block-uniform (scalar-loaded).
__global__ __launch_bounds__(256) void se_scale(const float* __restrict__ x,
                                                const float* __restrict__ g,
                                                float* __restrict__ out) {
    const int bc = blockIdx.x;
    const float gv = g[bc];
    const float4* xv = (const float4*)(x + (size_t)bc * SE_HW);
    float4* ov = (float4*)(out + (size_t)bc * SE_HW);
    const int t = threadIdx.x;

#pragma unroll
    for (int i = 0; i < 3; ++i) {
        float4 v = xv[t + i * 256];
        v.x *= gv; v.y *= gv; v.z *= gv; v.w *= gv;
        ov[t + i * 256] = v;
    }
    if (t < SE_HW4 - 768) {
        float4 v = xv[t + 768];
        v.x *= gv; v.y *= gv; v.z *= gv; v.w *= gv;
        ov[t + 768] = v;
    }
}

extern "C" void kernel_launch(void* const* d_in, const int* in_sizes, int n_in,
                              void* d_out, int out_size, void* d_ws, size_t ws_size,
                              hipStream_t stream) {
    const float* x  = (const float*)d_in[0];   // [32,256,56,56]
    const float* w1 = (const float*)d_in[1];   // [16,256]
    const float* b1 = (const float*)d_in[2];   // [16]
    const float* w2 = (const float*)d_in[3];   // [256,16]
    const float* b2 = (const float*)d_in[4];   // [256]
    float* out = (float*)d_out;

    float* s = (float*)d_ws;                   // [32*256] squeeze result
    float* g = s + SE_B * SE_C;                // [32*256] gate

    se_reduce<<<SE_B * SE_C, 256, 0, stream>>>(x, s);
    se_excite<<<1, 512, 0, stream>>>(s, w1, b1, w2, b2, g);
    se_scale<<<SE_B * SE_C, 256, 0, stream>>>(x, g, out);
}